// ConRoStage2Loss_38130719654041
// MI455X (gfx1250) — compile-verified
//
#include <hip/hip_runtime.h>
#include <math.h>

typedef __attribute__((ext_vector_type(2))) float v2f;
typedef __attribute__((ext_vector_type(8))) float v8f;

#define B_N 4096
#define D_K 256
#define SK_N 64   // S + K context vectors per row

// ---------------------------------------------------------------------------
// Kernel 1: per-row dots vs the 64 private (similar+diverse) vectors.
// Pure streaming pass over 256 MB -> bandwidth bound, plain FMAs are optimal.
// ---------------------------------------------------------------------------
__global__ __launch_bounds__(256) void sd_pass(
    const float* __restrict__ emb,
    const float* __restrict__ sim,
    const float* __restrict__ dvs,
    float* __restrict__ sd_sum,
    float* __restrict__ sd_exp)
{
    __shared__ float e[D_K];
    __shared__ float dots[SK_N];
    __shared__ float rs[SK_N];
    __shared__ float re[SK_N];

    const int i   = blockIdx.x;
    const int tid = threadIdx.x;              // 256 threads

    e[tid] = emb[(size_t)i * D_K + tid];
    __syncthreads();

    const int v    = tid >> 2;                // vector 0..63
    const int part = tid & 3;                 // 4 threads per vector
    const float* src = (v < 32)
        ? (sim + ((size_t)i * 32 + v) * D_K)
        : (dvs + ((size_t)i * 32 + (v - 32)) * D_K);

    const int d0 = part * 64;
    float p = 0.0f;
#pragma unroll 16
    for (int d = 0; d < 64; ++d)
        p = fmaf(e[d0 + d], src[d0 + d], p);

    // reduce the 4 partials (adjacent lanes)
    p += __shfl_xor(p, 1, 32);
    p += __shfl_xor(p, 2, 32);
    if (part == 0) dots[v] = p;
    __syncthreads();

    if (tid < SK_N) {
        const float dv = dots[tid];
        rs[tid] = dv;
        re[tid] = expf(dv);
    }
    __syncthreads();
    for (int off = 32; off > 0; off >>= 1) {
        if (tid < off) { rs[tid] += rs[tid + off]; re[tid] += re[tid + off]; }
        __syncthreads();
    }
    if (tid == 0) { sd_sum[i] = rs[0]; sd_exp[i] = re[0]; }
}

// ---------------------------------------------------------------------------
// Kernel 2: fused E*E^T with fp32 WMMA (V_WMMA_F32_16X16X4_F32) + row-wise
// masked reductions. One block = 16 rows x all 4096 columns; 8 waves split
// the 256 column tiles. The sims matrix is never materialized.
//
// fp32 WMMA fragment layout (ISA 7.12.2):
//   A (16x4):  lanes 0-15 -> M=lane, K={0,1}; lanes 16-31 -> M=lane-16, K={2,3}
//   B (4x16):  lanes 0-15 -> N=lane, K={0,1}; lanes 16-31 -> N=lane-16, K={2,3}
//   C/D vgpr e: lanes 0-15 -> M=e, N=lane; lanes 16-31 -> M=e+8, N=lane-16
// ---------------------------------------------------------------------------
__global__ __launch_bounds__(256) void pairs_pass(
    const float* __restrict__ emb,
    const int*   __restrict__ labels,
    float* __restrict__ rowexp,
    float* __restrict__ rowsum,
    float* __restrict__ rowcnt)
{
    const int tid  = threadIdx.x;
    const int wave = tid >> 5;                // 0..7
    const int lane = tid & 31;
    const int half = lane >> 4;               // 0 | 1
    const int lr   = lane & 15;
    const int r0   = blockIdx.x * 16;

    // Preload the 16x256 A-panel as 64 K4 fragments (held in VGPRs, reused
    // across all 32 column tiles of this wave).
    v2f afr[64];
    const float* arow = emb + (size_t)(r0 + lr) * D_K + 2 * half;
#pragma unroll
    for (int c = 0; c < 64; ++c)
        afr[c] = *(const v2f*)(arow + 4 * c);

    float expacc[8], sumacc[8], cntacc[8];
#pragma unroll
    for (int e = 0; e < 8; ++e) { expacc[e] = 0.f; sumacc[e] = 0.f; cntacc[e] = 0.f; }

    for (int t = 0; t < 32; ++t) {
        const int c0 = (t * 8 + wave) * 16;
        const float* brow = emb + (size_t)(c0 + lr) * D_K + 2 * half;

        v8f acc = {};
#pragma unroll
        for (int c = 0; c < 64; ++c) {
            const v2f b = *(const v2f*)(brow + 4 * c);
            acc = __builtin_amdgcn_wmma_f32_16x16x4_f32(
                false, afr[c], false, b, (short)0, acc, false, false);
        }

        const int  j    = c0 + lr;            // column this lane's elements use
        const bool malj = (labels[j] == 1);

#pragma unroll
        for (int e = 0; e < 8; ++e) {
            const int   r = r0 + e + 8 * half;
            const float v = acc[e];
            if (j != r) {
                expacc[e] += expf(v);
                if (malj) { sumacc[e] += v; cntacc[e] += 1.0f; }
            }
        }
    }

    // Butterfly-reduce across the 16 N-lanes of each half-wave.
#pragma unroll
    for (int e = 0; e < 8; ++e) {
        for (int m = 8; m >= 1; m >>= 1) {
            expacc[e] += __shfl_xor(expacc[e], m, 32);
            sumacc[e] += __shfl_xor(sumacc[e], m, 32);
            cntacc[e] += __shfl_xor(cntacc[e], m, 32);
        }
    }

    // Cross-wave reduce via LDS; each of the 16 rows written exactly once.
    __shared__ float red[3][8][16];           // [val][wave][local row]
    if (lr == 0) {
#pragma unroll
        for (int e = 0; e < 8; ++e) {
            const int rowL = e + 8 * half;
            red[0][wave][rowL] = expacc[e];
            red[1][wave][rowL] = sumacc[e];
            red[2][wave][rowL] = cntacc[e];
        }
    }
    __syncthreads();
    if (tid < 16) {
        float se = 0.f, ss = 0.f, sc = 0.f;
#pragma unroll
        for (int w = 0; w < 8; ++w) {
            se += red[0][w][tid];
            ss += red[1][w][tid];
            sc += red[2][w][tid];
        }
        rowexp[r0 + tid] = se;
        rowsum[r0 + tid] = ss;
        rowcnt[r0 + tid] = sc;
    }
}

// ---------------------------------------------------------------------------
// Kernel 3: combine per-row terms into the scalar loss.
// ---------------------------------------------------------------------------
__global__ __launch_bounds__(256) void finalize(
    const int*   __restrict__ labels,
    const float* __restrict__ rowexp,
    const float* __restrict__ rowsum,
    const float* __restrict__ rowcnt,
    const float* __restrict__ sd_sum,
    const float* __restrict__ sd_exp,
    float* __restrict__ out)
{
    __shared__ float acc[256];
    const int tid = threadIdx.x;
    float s = 0.0f;
    for (int i = tid; i < B_N; i += 256) {
        if (labels[i] == 1) {
            const float denom = rowexp[i] + sd_exp[i];
            const float num   = rowsum[i] + sd_sum[i];
            const float cnt   = rowcnt[i] + (float)SK_N;
            s += -num / cnt + logf(denom);
        }
    }
    acc[tid] = s;
    __syncthreads();
    for (int off = 128; off > 0; off >>= 1) {
        if (tid < off) acc[tid] += acc[tid + off];
        __syncthreads();
    }
    if (tid == 0) out[0] = acc[0] / (float)B_N;
}

// ---------------------------------------------------------------------------
extern "C" void kernel_launch(void* const* d_in, const int* in_sizes, int n_in,
                              void* d_out, int out_size, void* d_ws, size_t ws_size,
                              hipStream_t stream)
{
    (void)in_sizes; (void)n_in; (void)out_size; (void)ws_size;

    const float* emb    = (const float*)d_in[0];   // [B, D]
    const float* sim    = (const float*)d_in[1];   // [B, S, D]
    const float* dvs    = (const float*)d_in[2];   // [B, K, D]
    const int*   labels = (const int*)  d_in[3];   // [B]

    float* ws     = (float*)d_ws;
    float* rowexp = ws + 0 * B_N;
    float* rowsum = ws + 1 * B_N;
    float* rowcnt = ws + 2 * B_N;
    float* sdsum  = ws + 3 * B_N;
    float* sdexp  = ws + 4 * B_N;

    sd_pass   <<<B_N,      256, 0, stream>>>(emb, sim, dvs, sdsum, sdexp);
    pairs_pass<<<B_N / 16, 256, 0, stream>>>(emb, labels, rowexp, rowsum, rowcnt);
    finalize  <<<1,        256, 0, stream>>>(labels, rowexp, rowsum, rowcnt,
                                             sdsum, sdexp, (float*)d_out);
}